// QuantumPatternGenerator_55473797595924
// MI455X (gfx1250) — compile-verified
//
#include <hip/hip_runtime.h>
#include <math.h>

// D = A(16x4) x B(4x16) + C outer-sum via V_WMMA_F32_16X16X4_F32.
typedef float v2f __attribute__((ext_vector_type(2)));
typedef float v8f __attribute__((ext_vector_type(8)));

#define QPG_PI 3.14159265358979323846f
#define LUT_W  112   // 7 WMMA tiles * 16 = 112; pad region [100,112) never read

__global__ __launch_bounds__(256) void QuantumPatternGenerator_55473797595924_kernel(
    const float* __restrict__ patterns,   // [5,100]
    const float* __restrict__ sparsity_p, // [1]
    const int*   __restrict__ rows_p,     // [1]
    const int*   __restrict__ cols_p,     // [1]
    float*       __restrict__ out,
    unsigned total)
{
    __shared__ float vv[LUT_W];            // weighted table values, clamped pad
    __shared__ float T[LUT_W * LUT_W];     // {0,1} LUT (pad entries written from clamped vv, never read)

    const int tid = threadIdx.x;
    const float    thr   = 1.0f - 2.0f * sparsity_p[0];
    const unsigned rowsu = (unsigned)rows_p[0];
    const unsigned colsu = (unsigned)cols_p[0];

    // ---- Phase 1: vv[a] = 0.5 * sum_n (1/(n+1)) * patterns[n][a] ----
    if (tid < LUT_W) {
        const int a = (tid < 100) ? tid : 99;   // clamp pad entries
        float s = patterns[0 * 100 + a];
        s += patterns[1 * 100 + a] * (1.0f / 2.0f);
        s += patterns[2 * 100 + a] * (1.0f / 3.0f);
        s += patterns[3 * 100 + a] * (1.0f / 4.0f);
        s += patterns[4 * 100 + a] * (1.0f / 5.0f);
        vv[tid] = 0.5f * s;
    }
    __syncthreads();

    // ---- Phase 2: T[m][n] = (sin(pi*(vv[m]+vv[n])) > thr) ----
    // val[m][n] = vv[m] + vv[n] is rank-2: [vx,1] x [1;vy] -> one WMMA per 16x16 tile.
    {
        const int lane = tid & 31;
        const int wave = tid >> 5;
        const int lo   = (lane < 16) ? 1 : 0;   // data lives in lanes 0-15 (K0,K1); K2,K3 = 0
        const int l16  = lane & 15;
        for (int t = wave; t < 49; t += 8) {    // uniform per-wave loop -> EXEC all ones at WMMA
            const int m0 = (t / 7) * 16;
            const int n0 = (t % 7) * 16;
            v2f a, b;
            a.x = lo ? vv[m0 + l16] : 0.0f;     // A[m][0] = vx[m]
            a.y = lo ? 1.0f         : 0.0f;     // A[m][1] = 1
            b.x = lo ? 1.0f         : 0.0f;     // B[0][n] = 1
            b.y = lo ? vv[n0 + l16] : 0.0f;     // B[1][n] = vy[n]
            v8f c = {};
            v8f d = __builtin_amdgcn_wmma_f32_16x16x4_f32(
                false, a, false, b, (short)0, c, false, false);
            // D layout: VGPR r, lanes 0-15 -> (M=m0+r, N=n0+lane); lanes 16-31 -> M=m0+r+8
            const int n  = n0 + l16;
            const int mh = m0 + (lo ? 0 : 8);
#pragma unroll
            for (int r = 0; r < 8; ++r) {       // unconditional stores into padded LUT
                T[(mh + r) * LUT_W + n] = (__sinf(QPG_PI * d[r]) > thr) ? 1.0f : 0.0f;
            }
        }
    }
    __syncthreads();

    // ---- Phase 3: stream output, one float4 (global_store_b128) per iteration ----
    // px = (int)(row * 99/rows) is bit-exact vs (row/rows)*99 for pow2 rows.
    const float sx = 99.0f / (float)rowsu;
    const float sy = 99.0f / (float)colsu;
    const unsigned stride = gridDim.x * blockDim.x;
    const unsigned groups = (total + 3u) >> 2;
    unsigned g = blockIdx.x * blockDim.x + tid;
    if (g < groups) {
        // one divide up front, then row/col maintained incrementally (stride is fixed)
        unsigned e   = g << 2;
        unsigned row = e / colsu;
        unsigned col = e - row * colsu;
        const unsigned stride4 = stride << 2;
        const unsigned qs = stride4 / colsu;
        const unsigned rs = stride4 - qs * colsu;   // rs < colsu
        for (; g < groups; g += stride) {
            e = g << 2;
            if ((col + 4u <= colsu) & (e + 4u <= total)) {
                const int base = LUT_W * (int)((float)row * sx);
                float4 o;
                o.x = T[base + (int)((float)(col     ) * sy)];  // LDS broadcast loads
                o.y = T[base + (int)((float)(col + 1u) * sy)];
                o.z = T[base + (int)((float)(col + 2u) * sy)];
                o.w = T[base + (int)((float)(col + 3u) * sy)];
                *reinterpret_cast<float4*>(out + e) = o;
            } else {                                // rare tail: per-element, exact divmod
                for (unsigned k = 0; k < 4u && (e + k) < total; ++k) {
                    const unsigned ee = e + k;
                    const unsigned r2 = ee / colsu;
                    const unsigned c2 = ee - r2 * colsu;
                    out[ee] = T[LUT_W * (int)((float)r2 * sx) + (int)((float)c2 * sy)];
                }
            }
            row += qs;
            col += rs;
            if (col >= colsu) { col -= colsu; ++row; }
        }
    }
}

extern "C" void kernel_launch(void* const* d_in, const int* in_sizes, int n_in,
                              void* d_out, int out_size, void* d_ws, size_t ws_size,
                              hipStream_t stream) {
    (void)in_sizes; (void)n_in; (void)d_ws; (void)ws_size;
    const float* patterns = (const float*)d_in[0];
    const float* sparsity = (const float*)d_in[1];
    const int*   rows     = (const int*)d_in[2];
    const int*   cols     = (const int*)d_in[3];
    float* out = (float*)d_out;

    const unsigned total  = (unsigned)out_size;
    const unsigned groups = (total + 3u) >> 2;
    unsigned blocks = (groups + 255u) >> 8;
    if (blocks > 2048u) blocks = 2048u;   // 16K waves: saturates HBM stores while keeping
    if (blocks == 0u)   blocks = 1u;      // redundant per-block LUT work ~1 us aggregate
    QuantumPatternGenerator_55473797595924_kernel<<<blocks, 256, 0, stream>>>(
        patterns, sparsity, rows, cols, out, total);
}